// Conv2d_47940424958603
// MI455X (gfx1250) — compile-verified
//
#include <hip/hip_runtime.h>

typedef float v2f __attribute__((ext_vector_type(2)));
typedef float v8f __attribute__((ext_vector_type(8)));
typedef float f4  __attribute__((ext_vector_type(4)));

#define C_IN  8
#define C_OUT 16
#define HW    1024          // 32*32
#define N_IN  8192          // C_IN*H*W
#define N_OUT 16384         // C_OUT*OH*OW
#define MCOLS 16385u        // N_OUT+1
#define MTOT  134242305u    // (N_IN+1)*(N_OUT+1) = 8193*16385

// ---------------------------------------------------------------------------
// Analytic value of M[r][c] (gather formulation of the scatter in the ref).
// r < N_IN : weight position iff ky=iy-oh+1, kx=ix-ow+1 both in [0,3).
// r == N_IN: bias row, and M[N_IN][N_OUT] = 1.0.
// ---------------------------------------------------------------------------
__device__ __forceinline__ float m_val(unsigned r, unsigned c,
                                       const float* __restrict__ w,
                                       const float* __restrict__ bias) {
    if (r < N_IN) {
        if (c >= N_OUT) return 0.0f;                 // last column (except bias row)
        int cin = r >> 10, iy = (r >> 5) & 31, ix = r & 31;
        int oc  = c >> 10, oh = (c >> 5) & 31, ow = c & 31;
        int ky = iy - oh + 1, kx = ix - ow + 1;
        if ((unsigned)ky < 3u && (unsigned)kx < 3u)
            return w[((oc * C_IN + cin) * 3 + ky) * 3 + kx];
        return 0.0f;
    }
    return (c < N_OUT) ? bias[c >> 10] : 1.0f;       // bias row / homogeneous 1
}

// ---------------------------------------------------------------------------
// Streaming one-pass fill of M: 4 elements per thread, 16B-aligned float4
// non-temporal stores (537 MB >> 192 MB L2 -> do not cache). This is the
// roofline-critical kernel: ~23 us at 23.3 TB/s, pure store bandwidth.
// ---------------------------------------------------------------------------
__global__ void fill_M_kernel(const float* __restrict__ w,
                              const float* __restrict__ bias,
                              float* __restrict__ M) {
    unsigned idx = (blockIdx.x * blockDim.x + threadIdx.x) * 4u;
    if (idx >= MTOT) return;
    unsigned r = idx / MCOLS;
    unsigned c = idx - r * MCOLS;
    if (idx + 4u <= MTOT) {
        f4 v;
#pragma unroll
        for (int k = 0; k < 4; ++k) {
            v[k] = m_val(r, c, w, bias);
            if (++c == MCOLS) { c = 0u; ++r; }       // at most one row wrap per quad
        }
        __builtin_nontemporal_store(v, (f4*)(M + idx));   // global_store_b128 ... nt
    } else {
        for (; idx < MTOT; ++idx) {                  // single tail element (M[-1][-1]=1)
            M[idx] = m_val(r, c, w, bias);
            if (++c == MCOLS) { c = 0u; ++r; }
        }
    }
}

// ---------------------------------------------------------------------------
// lo/hi interval bounds as two fused 16x72x1024 GEMMs on V_WMMA_F32_16X16X4_F32.
//   lo = pos(W) @ im2col(l) + neg(W) @ im2col(u) + bias
//   hi = pos(W) @ im2col(u) + neg(W) @ im2col(l) + bias
// One wave per 16(M=oc) x 16(N=pixels) tile; K=72 fully unrolled (18 steps of 4,
// 72 WMMAs). Bias is pre-folded into the accumulator (C/D layout: vgpr r holds
// M = r + 8*half, broadcast over the 16 N lanes). All loads are unconditional
// (clamped address + cndmask) so EXEC stays all-ones through the whole kernel,
// per the WMMA restriction.
// ---------------------------------------------------------------------------
__global__ void bounds_wmma_kernel(const float* __restrict__ l,
                                   const float* __restrict__ u,
                                   const float* __restrict__ w,
                                   const float* __restrict__ bias,
                                   float* __restrict__ lo,
                                   float* __restrict__ hi) {
    const int lane = threadIdx.x;        // 0..31, one wave per block
    const int lid  = lane & 15;
    const int half = lane >> 4;
    const int tile = blockIdx.x;         // 0..63 tiles of 16 pixels
    const int oh   = tile >> 1;          // 32 pixels per row = 2 tiles
    const int ow   = (tile & 1) * 16 + lid;   // this lane's N column pixel

    // Fold bias into the accumulators: element (M,N) lives in vgpr (M&7) of
    // lanes [16*(M>=8) .. +15]; every N column gets the same bias[M].
    v8f acc_lo, acc_hi;
#pragma unroll
    for (int r = 0; r < 8; ++r) {
        const float b = bias[r + half * 8];
        acc_lo[r] = b;
        acc_hi[r] = b;
    }

#pragma unroll
    for (int k0 = 0; k0 < 72; k0 += 4) {
        const int kk = k0 + half * 2;    // this lane's two K indices: kk, kk+1

        // A fragments: rows oc = lid of pos(W)/neg(W), K = kk..kk+1
        const float w0 = w[lid * 72 + kk];
        const float w1 = w[lid * 72 + kk + 1];
        v2f ap, an;
        ap[0] = fmaxf(w0, 0.0f); ap[1] = fmaxf(w1, 0.0f);
        an[0] = fminf(w0, 0.0f); an[1] = fminf(w1, 0.0f);

        // B fragments: zero-padded im2col of l and u, col N = lid, K = kk..kk+1.
        // Branch-free: clamp the address, load unconditionally, select zero.
        v2f bl, bu;
#pragma unroll
        for (int j = 0; j < 2; ++j) {
            const int k   = kk + j;
            const int cin = k / 9;
            const int rem = k - cin * 9;
            const int ky  = rem / 3;
            const int kx  = rem - ky * 3;
            const int iy  = oh - 1 + ky;
            const int ix  = ow - 1 + kx;
            const bool valid = ((unsigned)iy < 32u) && ((unsigned)ix < 32u);
            const int  off   = valid ? (cin * HW + iy * 32 + ix) : 0;
            const float lv = l[off];
            const float uv = u[off];
            bl[j] = valid ? lv : 0.0f;
            bu[j] = valid ? uv : 0.0f;
        }

        acc_lo = __builtin_amdgcn_wmma_f32_16x16x4_f32(false, ap, false, bl,
                                                       (short)0, acc_lo, false, false);
        acc_lo = __builtin_amdgcn_wmma_f32_16x16x4_f32(false, an, false, bu,
                                                       (short)0, acc_lo, false, false);
        acc_hi = __builtin_amdgcn_wmma_f32_16x16x4_f32(false, ap, false, bu,
                                                       (short)0, acc_hi, false, false);
        acc_hi = __builtin_amdgcn_wmma_f32_16x16x4_f32(false, an, false, bl,
                                                       (short)0, acc_hi, false, false);
    }

    // Store per C/D layout: vgpr r holds M = r + half*8, N = lid.
    const int p = tile * 16 + lid;       // pixel index within (OH*OW)
#pragma unroll
    for (int r = 0; r < 8; ++r) {
        const int oc = r + half * 8;
        lo[oc * HW + p] = acc_lo[r];
        hi[oc * HW + p] = acc_hi[r];
    }
}

// ---------------------------------------------------------------------------
extern "C" void kernel_launch(void* const* d_in, const int* in_sizes, int n_in,
                              void* d_out, int out_size, void* d_ws, size_t ws_size,
                              hipStream_t stream) {
    const float* l    = (const float*)d_in[0];   // concrete_lower (8192)
    const float* u    = (const float*)d_in[1];   // concrete_upper (8192)
    const float* w    = (const float*)d_in[2];   // weights (16,8,3,3)
    const float* bias = (const float*)d_in[3];   // bias (16)

    float* lo = (float*)d_out;                   // 16384
    float* hi = lo + N_OUT;                      // 16384
    float* M  = hi + N_OUT;                      // 8193*16385 (16B-aligned: 32768 floats in)

    // Tiny WMMA GEMM for the bounds (64 waves).
    bounds_wmma_kernel<<<64, 32, 0, stream>>>(l, u, w, bias, lo, hi);

    // Bandwidth-bound streaming fill of M.
    const unsigned units  = (MTOT + 3u) / 4u;            // float4 units (+1-elem tail)
    const unsigned blocks = (units + 255u) / 256u;
    fill_M_kernel<<<blocks, 256, 0, stream>>>(w, bias, M);
}